// Attention_54099408060483
// MI455X (gfx1250) — compile-verified
//
#include <hip/hip_runtime.h>
#include <math.h>

typedef __attribute__((ext_vector_type(2))) float v2f;
typedef __attribute__((ext_vector_type(8))) float v8f;
typedef int vi4 __attribute__((vector_size(16)));   // <4 x i32> for async-LDS builtin

#if defined(__has_builtin)
#if __has_builtin(__builtin_amdgcn_global_load_async_to_lds_b128)
#define HAVE_ASYNC_LDS 1
#endif
#endif

namespace {
constexpr int Bc = 16;     // batch
constexpr int Sc = 2048;   // sequence
constexpr int Dc = 128;    // head dim
constexpr int QT = 64;     // q rows per block (4 waves x 16)
constexpr int KT = 16;     // key-tile length (double buffered)
constexpr int NW = 4;      // waves per block
constexpr int NTHR = 128;  // threads per block
constexpr int NKT = Sc / KT;                       // 128 key tiles
constexpr int KCQK = Dc / 4;                       // 32 K=4 steps over D
constexpr int KCPV = KT / 4;                       // 4  K=4 steps over key tile
constexpr int F4_PER_THR = (KT * Dc / 4) / NTHR;   // 4 float4 per thread per tile
constexpr float SCALE = 0.08838834764831845f;      // 1/sqrt(128)
}

__device__ __forceinline__ void wait_async_zero() {
#if defined(__HIP_DEVICE_COMPILE__)
  asm volatile("s_wait_asynccnt 0x0" ::: "memory");
#endif
}

// Stream one KT x Dc fp32 tile into LDS. Async path: ASYNCcnt-tracked DMA,
// no VGPR staging. Fallback: plain b128 load/store pair.
__device__ __forceinline__ void tile_to_lds(const float* __restrict__ g,
                                            float* l, int tid) {
#pragma unroll
  for (int i = 0; i < F4_PER_THR; ++i) {
    const float* gs = g + (size_t)(tid + i * NTHR) * 4;
    float* ls = l + (tid + i * NTHR) * 4;
#if defined(HAVE_ASYNC_LDS) && defined(__HIP_DEVICE_COMPILE__)
    __builtin_amdgcn_global_load_async_to_lds_b128(
        (__attribute__((address_space(1))) vi4*)(float*)gs,
        (__attribute__((address_space(3))) vi4*)ls, 0, 0);
#else
    *(float4*)ls = *(const float4*)gs;
#endif
  }
}

__global__ __launch_bounds__(NTHR, 1)
void attn_fused(const float* __restrict__ Q, const float* __restrict__ K,
                const float* __restrict__ V, float* __restrict__ Out,
                float* __restrict__ W) {
  __shared__ float Klds[2][KT][Dc];    // 16 KB, double buffered
  __shared__ float Vlds[2][KT][Dc];    // 16 KB, double buffered
  __shared__ float Plds[NW][16][KT];   //  4 KB, per-wave private

  const int tid   = threadIdx.x;
  const int wave  = tid >> 5;
  const int lane  = tid & 31;
  const int lhalf = lane >> 4;   // 0..1 (lane half)
  const int l16   = lane & 15;   // 0..15

  const int b  = blockIdx.x / (Sc / QT);
  const int qt = blockIdx.x % (Sc / QT);
  const int m0 = qt * QT + wave * 16;    // first q row owned by this wave

  const float* Qb = Q + (size_t)b * Sc * Dc;
  const float* Kb = K + (size_t)b * Sc * Dc;
  const float* Vb = V + (size_t)b * Sc * Dc;
  float* Outb = Out + (size_t)b * Sc * Dc;
  float* Wb   = W   + (size_t)b * Sc * Sc;

  // ---- Q A-fragments, resident in VGPRs for the whole kernel ----
  // A layout (32-bit, 16x4): lane L -> row M = L%16, K = 2*(L/16)+{0,1}
  v2f qf[KCQK];
  {
    const v2f* q2 = (const v2f*)(Qb + (size_t)(m0 + l16) * Dc + 2 * lhalf);
#pragma unroll
    for (int kc = 0; kc < KCQK; ++kc) qf[kc] = q2[2 * kc];
  }

  float rmax[8], rsum[8];
#pragma unroll
  for (int r = 0; r < 8; ++r) { rmax[r] = -INFINITY; rsum[r] = 0.0f; }

  // ---------------- pass 1: online row max / sum over all keys ----------------
  tile_to_lds(Kb, &Klds[0][0][0], tid);
  for (int kt = 0; kt < NKT; ++kt) {
    const int cur = kt & 1;
    wait_async_zero();     // my slice of the current tile has landed
    __syncthreads();       // everyone's slice landed; previous buffer released
    if (kt + 1 < NKT)
      tile_to_lds(Kb + (size_t)(kt + 1) * KT * Dc, &Klds[cur ^ 1][0][0], tid);

    v8f acc = {};
#pragma unroll
    for (int kc = 0; kc < KCQK; ++kc) {
      // B layout (32-bit, 4x16): lane L -> col N = L%16, K = 2*(L/16)+{0,1}
      const v2f bf = *(const v2f*)&Klds[cur][l16][4 * kc + 2 * lhalf];
      acc = __builtin_amdgcn_wmma_f32_16x16x4_f32(
          false, qf[kc], false, bf, (short)0, acc, false, false);
    }

    // C layout: element r -> row M = r + 8*lhalf, col N = l16
#pragma unroll
    for (int r = 0; r < 8; ++r) {
      float tmax = acc[r] * SCALE;
#pragma unroll
      for (int off = 1; off < 16; off <<= 1)
        tmax = fmaxf(tmax, __shfl_xor(tmax, off, 32));
      const float nmax = fmaxf(rmax[r], tmax);
      const float corr = __expf(rmax[r] - nmax);
      float psum = __expf(acc[r] * SCALE - nmax);
#pragma unroll
      for (int off = 1; off < 16; off <<= 1)
        psum += __shfl_xor(psum, off, 32);
      rsum[r] = rsum[r] * corr + psum;
      rmax[r] = nmax;
    }
  }

  float rinv[8];
#pragma unroll
  for (int r = 0; r < 8; ++r) rinv[r] = 1.0f / rsum[r];

  v8f oacc[Dc / 16];
#pragma unroll
  for (int dt = 0; dt < Dc / 16; ++dt) oacc[dt] = (v8f){};

  // -------- pass 2: recompute scores, write weights, accumulate P*V --------
  tile_to_lds(Kb, &Klds[0][0][0], tid);
  tile_to_lds(Vb, &Vlds[0][0][0], tid);
  for (int kt = 0; kt < NKT; ++kt) {
    const int cur = kt & 1;
    wait_async_zero();
    __syncthreads();
    if (kt + 1 < NKT) {
      tile_to_lds(Kb + (size_t)(kt + 1) * KT * Dc, &Klds[cur ^ 1][0][0], tid);
      tile_to_lds(Vb + (size_t)(kt + 1) * KT * Dc, &Vlds[cur ^ 1][0][0], tid);
    }

    v8f acc = {};
#pragma unroll
    for (int kc = 0; kc < KCQK; ++kc) {
      const v2f bf = *(const v2f*)&Klds[cur][l16][4 * kc + 2 * lhalf];
      acc = __builtin_amdgcn_wmma_f32_16x16x4_f32(
          false, qf[kc], false, bf, (short)0, acc, false, false);
    }

    // normalized probabilities -> streaming global store + LDS stash (A-layout src)
#pragma unroll
    for (int r = 0; r < 8; ++r) {
      const float p = __expf(acc[r] * SCALE - rmax[r]) * rinv[r];
      const int mrow = r + 8 * lhalf;
      __builtin_nontemporal_store(
          p, &Wb[(size_t)(m0 + mrow) * Sc + (size_t)kt * KT + l16]);
      Plds[wave][mrow][l16] = p;
    }

    // P (16 x KT) @ V (KT x 128); per-wave LDS P, DS pipe is in-order per wave
#pragma unroll
    for (int dt = 0; dt < Dc / 16; ++dt) {
#pragma unroll
      for (int kc = 0; kc < KCPV; ++kc) {
        const v2f a = *(const v2f*)&Plds[wave][l16][4 * kc + 2 * lhalf];
        v2f bf;
        bf.x = Vlds[cur][4 * kc + 2 * lhalf][dt * 16 + l16];
        bf.y = Vlds[cur][4 * kc + 2 * lhalf + 1][dt * 16 + l16];
        oacc[dt] = __builtin_amdgcn_wmma_f32_16x16x4_f32(
            false, a, false, bf, (short)0, oacc[dt], false, false);
      }
    }
  }

  // ---- write out ----
#pragma unroll
  for (int dt = 0; dt < Dc / 16; ++dt) {
#pragma unroll
    for (int r = 0; r < 8; ++r) {
      Outb[(size_t)(m0 + r + 8 * lhalf) * Dc + dt * 16 + l16] = oacc[dt][r];
    }
  }
}

extern "C" void kernel_launch(void* const* d_in, const int* in_sizes, int n_in,
                              void* d_out, int out_size, void* d_ws, size_t ws_size,
                              hipStream_t stream) {
  (void)in_sizes; (void)n_in; (void)d_ws; (void)ws_size; (void)out_size;
  const float* Q = (const float*)d_in[0];
  const float* K = (const float*)d_in[1];
  const float* V = (const float*)d_in[2];
  float* Out = (float*)d_out;                 // [B,S,D] first
  float* W   = Out + (size_t)Bc * Sc * Dc;    // then [B,S,S]
  dim3 grid(Bc * (Sc / QT));   // 512 blocks
  dim3 block(NTHR);            // 128 threads = 4 wave32
  attn_fused<<<grid, block, 0, stream>>>(Q, K, V, Out, W);
}